// CNODExtmod_14087492731659
// MI455X (gfx1250) — compile-verified
//
#include <hip/hip_runtime.h>

// ---------------- problem constants (match reference) ----------------
#define B_TOTAL   4096
#define T_STEPS   100
#define HID       128
#define DIM       65        // 2*NC + 1
#define DIMP      68        // DIM padded up to multiple of 4 (WMMA K granularity)
#define OUTP      80        // DIM padded up to multiple of 16 (WMMA N granularity)
#define DT        0.25f
#define EULER     4

// ---------------- launch configuration ----------------
#define ROWS_PER_WAVE 16
#define WAVES_PER_WG  2
#define ROWS_PER_WG   (ROWS_PER_WAVE * WAVES_PER_WG)
#define NTHREADS      (WAVES_PER_WG * 32)          // wave32

// ---------------- LDS layout (floats) ----------------
// Weights are stored K-quad interleaved: element (k,n) -> ((k>>2)*N + n)*4 + (k&3)
// so each WMMA B fragment {W[k][n], W[k+1][n]} is a contiguous, 8B-aligned pair
// (single ds_load_b64, no VGPR repacking) and the two EXEC-halves of the wave
// hit disjoint bank sets (conflict-free across 64 banks).
#define SZ_W1 (DIMP * HID)               // 8704   (K-padded, rows 65..67 zero)
#define SZ_W2 (HID * HID)                // 16384
#define SZ_W3 (HID * OUTP)               // 10240  (N-padded, cols 65..79 zero)
#define SZ_H  (ROWS_PER_WAVE * DIMP)     // 1088 per wave
#define SZ_A  (ROWS_PER_WAVE * HID)      // 2048 per wave
#define WAVE_FLOATS (SZ_H + 2 * SZ_A)
#define LDS_FLOATS  (SZ_W1 + SZ_W2 + SZ_W3 + HID + HID + OUTP + WAVES_PER_WG * WAVE_FLOATS)

typedef float v2f __attribute__((ext_vector_type(2)));
typedef float v8f __attribute__((ext_vector_type(8)));

// Order cross-lane LDS producer->consumer within this wave (DS ops are
// in-order per wave; the compiler cannot see cross-lane deps, so force it).
#define LDS_FENCE() asm volatile("s_wait_dscnt 0" ::: "memory")

#define V8F_ZERO {0.f, 0.f, 0.f, 0.f, 0.f, 0.f, 0.f, 0.f}

__device__ __forceinline__ v8f wmma_f32(v2f a, v2f b, v8f c) {
  // V_WMMA_F32_16X16X4_F32: (neg_a, A, neg_b, B, c_mod, C, reuse_a, reuse_b)
  return __builtin_amdgcn_wmma_f32_16x16x4_f32(false, a, false, b, (short)0, c,
                                               false, false);
}

// Four N-tiles sharing one A fragment per K-step: four independent WMMA
// accumulation chains so ~4 WMMA issues of compute cover each round of LDS
// load latency, and A-side DS traffic is quartered.
template <int KTOT>
__device__ __forceinline__ void gemm_quad(const float* __restrict__ A, int apitch,
                                          const float* __restrict__ W, int wpitch,
                                          int n0, int arow, int col, int khalf,
                                          v8f& acc0, v8f& acc1, v8f& acc2,
                                          v8f& acc3) {
#pragma unroll
  for (int k0 = 0; k0 < KTOT; k0 += 4) {
    v2f a = *(const v2f*)(A + arow * apitch + k0 + khalf);
    const float* wp = W + k0 * wpitch + ((n0 + col) << 2) + khalf;
    v2f b0 = *(const v2f*)(wp);
    v2f b1 = *(const v2f*)(wp + 64);    // +16 cols * 4 interleave
    v2f b2 = *(const v2f*)(wp + 128);
    v2f b3 = *(const v2f*)(wp + 192);
    acc0 = wmma_f32(a, b0, acc0);
    acc1 = wmma_f32(a, b1, acc1);
    acc2 = wmma_f32(a, b2, acc2);
    acc3 = wmma_f32(a, b3, acc3);
  }
}

template <int KTOT>
__device__ __forceinline__ void gemm_one(const float* __restrict__ A, int apitch,
                                         const float* __restrict__ W, int wpitch,
                                         int n0, int arow, int col, int khalf,
                                         v8f& acc) {
#pragma unroll
  for (int k0 = 0; k0 < KTOT; k0 += 4) {
    v2f a = *(const v2f*)(A + arow * apitch + k0 + khalf);
    const float* wp = W + k0 * wpitch + ((n0 + col) << 2) + khalf;
    v2f b = *(const v2f*)(wp);
    acc = wmma_f32(a, b, acc);
  }
}

__device__ __forceinline__ void store_relu(float* __restrict__ dst, int pitch,
                                           int n0, int col, int rbase,
                                           const v8f& acc, float bias) {
#pragma unroll
  for (int i = 0; i < 8; ++i) {
    float v = acc[i] + bias;
    dst[(rbase + i) * pitch + n0 + col] = v > 0.f ? v : 0.f;
  }
}

__global__ __launch_bounds__(NTHREADS, 1) void cnode_persistent_kernel(
    const float* __restrict__ times, const float* __restrict__ Y,
    const float* __restrict__ mask,
    const float* __restrict__ W1, const float* __restrict__ b1,
    const float* __restrict__ W2, const float* __restrict__ b2,
    const float* __restrict__ W3, const float* __restrict__ b3,
    float* __restrict__ out_ypred, float* __restrict__ out_ytraj,
    float* __restrict__ out_times, float* __restrict__ out_h,
    float* __restrict__ out_hcn) {
  extern __shared__ float smem[];
  const int tid  = threadIdx.x;
  const int lane = tid & 31;
  const int wave = tid >> 5;

  float* sW1 = smem;                 // [DIMP x HID]  K-quad interleaved
  float* sW2 = sW1 + SZ_W1;          // [HID  x HID]  K-quad interleaved
  float* sW3 = sW2 + SZ_W2;          // [HID  x OUTP] K-quad interleaved
  float* sb1 = sW3 + SZ_W3;          // [HID]
  float* sb2 = sb1 + HID;            // [HID]
  float* sb3 = sb2 + HID;            // [OUTP]
  float* sh  = sb3 + OUTP + wave * WAVE_FLOATS;  // [16 x DIMP] state h
  float* sa1 = sh + SZ_H;                        // [16 x HID]
  float* sa2 = sa1 + SZ_A;                       // [16 x HID]

  // ---- cooperative weight staging, K-quad interleave (once per launch) ----
  for (int i = tid; i < SZ_W1; i += NTHREADS) {
    int k = i >> 7, n = i & (HID - 1);
    float v = (k < DIM) ? W1[k * HID + n] : 0.f;
    sW1[(((k >> 2) * HID + n) << 2) | (k & 3)] = v;
  }
  for (int i = tid; i < SZ_W2; i += NTHREADS) {
    int k = i >> 7, n = i & (HID - 1);
    sW2[(((k >> 2) * HID + n) << 2) | (k & 3)] = W2[i];
  }
  for (int i = tid; i < SZ_W3; i += NTHREADS) {
    int k = i / OUTP, n = i - k * OUTP;
    float v = (n < DIM) ? W3[k * DIM + n] : 0.f;
    sW3[(((k >> 2) * OUTP + n) << 2) | (k & 3)] = v;
  }
  for (int i = tid; i < HID; i += NTHREADS) { sb1[i] = b1[i]; sb2[i] = b2[i]; }
  for (int i = tid; i < OUTP; i += NTHREADS) sb3[i] = (i < DIM) ? b3[i] : 0.f;
  // h0 = 0 (padded cols 65..67 stay zero forever)
  for (int i = lane; i < SZ_H; i += 32) sh[i] = 0.f;

  if (blockIdx.x == 0)
    for (int i = tid; i < T_STEPS; i += NTHREADS) out_times[i] = times[i];

  __syncthreads();

  const int rowBase = blockIdx.x * ROWS_PER_WG + wave * ROWS_PER_WAVE;
  const int col   = lane & 15;         // N column within tile (B/C/D frags)
  const int arow  = lane & 15;         // M row (A frag)
  const int khalf = (lane >> 4) << 1;  // lanes 16..31 hold K+2,K+3
  const int rbase = (lane >> 4) << 3;  // C/D: lanes 16..31 hold rows 8..15

  for (int t = 0; t < T_STEPS; ++t) {
    for (int s = 0; s < EULER; ++s) {
      // ---------- layer 1: a1 = relu(h @ W1 + b1), K=68(65) ----------
      for (int np = 0; np < HID / 64; ++np) {
        const int n0 = np * 64;
        v8f acc0 = V8F_ZERO, acc1 = V8F_ZERO, acc2 = V8F_ZERO, acc3 = V8F_ZERO;
        gemm_quad<DIMP>(sh, DIMP, sW1, HID, n0, arow, col, khalf,
                        acc0, acc1, acc2, acc3);
        store_relu(sa1, HID, n0,      col, rbase, acc0, sb1[n0 + col]);
        store_relu(sa1, HID, n0 + 16, col, rbase, acc1, sb1[n0 + 16 + col]);
        store_relu(sa1, HID, n0 + 32, col, rbase, acc2, sb1[n0 + 32 + col]);
        store_relu(sa1, HID, n0 + 48, col, rbase, acc3, sb1[n0 + 48 + col]);
      }
      LDS_FENCE();
      // ---------- layer 2: a2 = relu(a1 @ W2 + b2), K=128 ----------
      for (int np = 0; np < HID / 64; ++np) {
        const int n0 = np * 64;
        v8f acc0 = V8F_ZERO, acc1 = V8F_ZERO, acc2 = V8F_ZERO, acc3 = V8F_ZERO;
        gemm_quad<HID>(sa1, HID, sW2, HID, n0, arow, col, khalf,
                       acc0, acc1, acc2, acc3);
        store_relu(sa2, HID, n0,      col, rbase, acc0, sb2[n0 + col]);
        store_relu(sa2, HID, n0 + 16, col, rbase, acc1, sb2[n0 + 16 + col]);
        store_relu(sa2, HID, n0 + 32, col, rbase, acc2, sb2[n0 + 32 + col]);
        store_relu(sa2, HID, n0 + 48, col, rbase, acc3, sb2[n0 + 48 + col]);
      }
      LDS_FENCE();
      // ---------- layer 3 + Euler: h += DT * (a2 @ W3 + b3) ----------
      // 5 N-tiles: one quad (cols 0..63, all real state) + one single (64..79)
      {
        v8f acc0 = V8F_ZERO, acc1 = V8F_ZERO, acc2 = V8F_ZERO, acc3 = V8F_ZERO;
        gemm_quad<HID>(sa2, HID, sW3, OUTP, 0, arow, col, khalf,
                       acc0, acc1, acc2, acc3);
        const float bias0 = sb3[col];
        const float bias1 = sb3[16 + col];
        const float bias2 = sb3[32 + col];
        const float bias3 = sb3[48 + col];
#pragma unroll
        for (int i = 0; i < 8; ++i) {
          float* hp = sh + (rbase + i) * DIMP + col;
          hp[0]  += DT * (acc0[i] + bias0);
          hp[16] += DT * (acc1[i] + bias1);
          hp[32] += DT * (acc2[i] + bias2);
          hp[48] += DT * (acc3[i] + bias3);
        }
      }
      {
        const int n0 = 64;
        v8f acc = V8F_ZERO;
        gemm_one<HID>(sa2, HID, sW3, OUTP, n0, arow, col, khalf, acc);
        if (n0 + col < DIM) {  // only col==0 maps to a real state element
          const float bias = sb3[n0 + col];
#pragma unroll
          for (int i = 0; i < 8; ++i) {
            float* hp = sh + (rbase + i) * DIMP + n0 + col;
            *hp += DT * (acc[i] + bias);
          }
        }
      }
      LDS_FENCE();
    }
    // ---------- observation step: one lane per row ----------
    if (lane < 16) {
      const int brow = rowBase + lane;
      const float y = Y[brow * T_STEPS + t];
      const float m = mask[brow * T_STEPS + t];
      float* hr = sh + lane * DIMP;
      const float ypred = hr[0];                     // h_cn[:, 0]
      out_ypred[brow * T_STEPS + t] = ypred;
      out_ytraj[brow * T_STEPS + t] = ypred;
      if (t == T_STEPS - 1) {
        for (int j = 0; j < DIM; ++j) out_hcn[brow * DIM + j] = hr[j];
      }
      // h_new = m * [y, h_cn[:-1]] + (1-m) * h_cn  (in-place, descending j)
      const float mi = 1.f - m;
      for (int j = DIM - 1; j >= 1; --j) hr[j] = m * hr[j - 1] + mi * hr[j];
      hr[0] = m * y + mi * ypred;
    }
    LDS_FENCE();
  }
  // ---------- final state ----------
  if (lane < 16) {
    const int brow = rowBase + lane;
    const float* hr = sh + lane * DIMP;
    for (int j = 0; j < DIM; ++j) out_h[brow * DIM + j] = hr[j];
  }
}

extern "C" void kernel_launch(void* const* d_in, const int* in_sizes, int n_in,
                              void* d_out, int out_size, void* d_ws,
                              size_t ws_size, hipStream_t stream) {
  const float* times = (const float*)d_in[0];
  const float* Y     = (const float*)d_in[1];
  const float* mask  = (const float*)d_in[2];
  const float* W1    = (const float*)d_in[3];
  const float* b1    = (const float*)d_in[4];
  const float* W2    = (const float*)d_in[5];
  const float* b2    = (const float*)d_in[6];
  const float* W3    = (const float*)d_in[7];
  const float* b3    = (const float*)d_in[8];

  float* out       = (float*)d_out;
  float* out_ypred = out;
  float* out_ytraj = out_ypred + (size_t)B_TOTAL * T_STEPS;
  float* out_times = out_ytraj + (size_t)B_TOTAL * T_STEPS;
  float* out_h     = out_times + T_STEPS;
  float* out_hcn   = out_h + (size_t)B_TOTAL * DIM;

  dim3 grid(B_TOTAL / ROWS_PER_WG);
  dim3 block(NTHREADS);
  size_t shmem = (size_t)LDS_FLOATS * sizeof(float);  // ~184 KB of 320 KB WGP LDS
  hipLaunchKernelGGL(cnode_persistent_kernel, grid, block, shmem, stream, times,
                     Y, mask, W1, b1, W2, b2, W3, b3, out_ypred, out_ytraj,
                     out_times, out_h, out_hcn);
}